// Top2Gate_15796889714906
// MI455X (gfx1250) — compile-verified
//
#include <hip/hip_runtime.h>

typedef __attribute__((ext_vector_type(2))) float v2f;
typedef __attribute__((ext_vector_type(4))) float v4f;
typedef __attribute__((ext_vector_type(8))) float v8f;

constexpr int T = 8192, D = 2048, E = 64;
constexpr int CAP = 256;               // 2 * ceil(T/E)
constexpr int NB = 32, CHUNK = 256;    // token blocking for location scan
constexpr int TB = T / 8;              // blocks in top2 kernel (8 tokens/block)
constexpr float EPSF = 1.1920929e-07f; // float32 eps

// ---------------------------------------------------------------- zero fill
__global__ void fill_zero4(v4f* __restrict__ p, long long n4,
                           float* __restrict__ tail, int ntail) {
  long long i = (long long)blockIdx.x * blockDim.x + threadIdx.x;
  const long long stride = (long long)gridDim.x * blockDim.x;
  v4f z = {0.f, 0.f, 0.f, 0.f};
  for (; i < n4; i += stride) p[i] = z;
  if (blockIdx.x == 0 && (int)threadIdx.x < ntail) tail[threadIdx.x] = 0.f;
}

// ---------------------------------------------------------------- GEMM (WMMA f32)
// logits[T,E] = x[T,D] @ w[E,D]^T.  Block = 4 waves; wave -> 16 M-rows, 4 N-tiles.
__global__ __launch_bounds__(128) void gemm_logits(const float* __restrict__ x,
                                                   const float* __restrict__ w,
                                                   float* __restrict__ logits) {
  const int lane = threadIdx.x & 31;
  const int wave = threadIdx.x >> 5;
  const int half = lane >> 4;   // K base selector (0 -> K0/K1, 1 -> K2/K3)
  const int l16  = lane & 15;
  const int m0 = blockIdx.x * 64 + wave * 16;

  const float* arow = x + (size_t)(m0 + l16) * D + 2 * half;
  const float* brow = w + (size_t)l16 * D + 2 * half;  // N-tile n adds n*16*D

  v8f acc0 = {}, acc1 = {}, acc2 = {}, acc3 = {};
  for (int k = 0; k < D; k += 4) {
    v2f a  = *(const v2f*)(arow + k);
    v2f b0 = *(const v2f*)(brow + k);
    v2f b1 = *(const v2f*)(brow + (size_t)16 * D + k);
    v2f b2 = *(const v2f*)(brow + (size_t)32 * D + k);
    v2f b3 = *(const v2f*)(brow + (size_t)48 * D + k);
    acc0 = __builtin_amdgcn_wmma_f32_16x16x4_f32(false, a, false, b0, (short)0, acc0, false, false);
    acc1 = __builtin_amdgcn_wmma_f32_16x16x4_f32(false, a, false, b1, (short)0, acc1, false, false);
    acc2 = __builtin_amdgcn_wmma_f32_16x16x4_f32(false, a, false, b2, (short)0, acc2, false, false);
    acc3 = __builtin_amdgcn_wmma_f32_16x16x4_f32(false, a, false, b3, (short)0, acc3, false, false);
  }
#pragma unroll
  for (int r = 0; r < 8; ++r) {
    const int m = m0 + r + 8 * half;
    float* row = logits + (size_t)m * E + l16;
    row[0]  = acc0[r];
    row[16] = acc1[r];
    row[32] = acc2[r];
    row[48] = acc3[r];
  }
}

// ---------------------------------------------------------------- softmax + top2
// One wave per token; lane owns experts (lane, lane+32).
__global__ __launch_bounds__(256) void top2_gate(const float* __restrict__ logits,
                                                 const float* __restrict__ gnoise,
                                                 int* __restrict__ idx1, int* __restrict__ idx2,
                                                 float* __restrict__ g1, float* __restrict__ g2,
                                                 float* __restrict__ gpart) {
  __shared__ float gs[8][E];
  const int lane = threadIdx.x & 31;
  const int wv   = threadIdx.x >> 5;
  const int t = blockIdx.x * 8 + wv;

  const float l0 = logits[(size_t)t * E + lane];
  const float l1 = logits[(size_t)t * E + lane + 32];

  float m = fmaxf(l0, l1);
  for (int off = 16; off; off >>= 1) m = fmaxf(m, __shfl_xor(m, off, 32));
  const float e0 = expf(l0 - m), e1 = expf(l1 - m);
  float s = e0 + e1;
  for (int off = 16; off; off >>= 1) s += __shfl_xor(s, off, 32);
  const float gg0 = e0 / s, gg1 = e1 / s;

  // argmax of logits (first-index tie break, matches jnp.argmax)
  float bv = l0; int bi = lane;
  if (l1 > bv) { bv = l1; bi = lane + 32; }
  for (int off = 16; off; off >>= 1) {
    float ov = __shfl_xor(bv, off, 32);
    int   oi = __shfl_xor(bi, off, 32);
    if (ov > bv || (ov == bv && oi < bi)) { bv = ov; bi = oi; }
  }
  const int i1 = bi;

  // second expert on gumbel-noised logits, expert i1 masked to -inf
  float n0 = l0 + gnoise[(size_t)t * E + lane];
  float n1 = l1 + gnoise[(size_t)t * E + lane + 32];
  if (lane == i1)      n0 = -__builtin_inff();
  if (lane + 32 == i1) n1 = -__builtin_inff();
  float cv = n0; int ci = lane;
  if (n1 > cv) { cv = n1; ci = lane + 32; }
  for (int off = 16; off; off >>= 1) {
    float ov = __shfl_xor(cv, off, 32);
    int   oi = __shfl_xor(ci, off, 32);
    if (ov > cv || (ov == cv && oi < ci)) { cv = ov; ci = oi; }
  }
  const int i2 = ci;

  // fetch gate values at i1, i2 via lane broadcast
  const float p0 = __shfl(gg0, i1 & 31, 32), p1 = __shfl(gg1, i1 & 31, 32);
  const float q0 = __shfl(gg0, i2 & 31, 32), q1 = __shfl(gg1, i2 & 31, 32);
  if (lane == 0) {
    idx1[t] = i1; idx2[t] = i2;
    g1[t] = (i1 < 32) ? p0 : p1;
    g2[t] = (i2 < 32) ? q0 : q1;
  }

  // deterministic per-block gate sums (for l_aux's `me`)
  gs[wv][lane]      = gg0;
  gs[wv][lane + 32] = gg1;
  __syncthreads();
  if (threadIdx.x < E) {
    float acc = 0.f;
    for (int w2 = 0; w2 < 8; ++w2) acc += gs[w2][threadIdx.x];
    gpart[(size_t)blockIdx.x * E + threadIdx.x] = acc;
  }
}

// ---------------------------------------------------------------- per-chunk expert histograms
__global__ __launch_bounds__(CHUNK) void hist_kernel(const int* __restrict__ idx1,
                                                     const int* __restrict__ idx2,
                                                     int* __restrict__ hist1,
                                                     int* __restrict__ hist2) {
  __shared__ int h1[E], h2[E];
  if (threadIdx.x < E) { h1[threadIdx.x] = 0; h2[threadIdx.x] = 0; }
  __syncthreads();
  const int t = blockIdx.x * CHUNK + threadIdx.x;
  atomicAdd(&h1[idx1[t]], 1);
  atomicAdd(&h2[idx2[t]], 1);
  __syncthreads();
  if (threadIdx.x < E) {
    hist1[blockIdx.x * E + threadIdx.x] = h1[threadIdx.x];
    hist2[blockIdx.x * E + threadIdx.x] = h2[threadIdx.x];
  }
}

// ---------------------------------------------------------------- cross-chunk scan + l_aux
__global__ __launch_bounds__(64) void scan_kernel(const int* __restrict__ hist1,
                                                  const int* __restrict__ hist2,
                                                  int* __restrict__ off1, int* __restrict__ off2,
                                                  int* __restrict__ count1,
                                                  const float* __restrict__ gpart,
                                                  float* __restrict__ out_laux) {
  __shared__ float part[E];
  const int e = threadIdx.x;
  int r1 = 0, r2 = 0;
  for (int b = 0; b < NB; ++b) {
    off1[b * E + e] = r1; r1 += hist1[b * E + e];
    off2[b * E + e] = r2; r2 += hist2[b * E + e];
  }
  count1[e] = r1;
  float gsum = 0.f;
  for (int b = 0; b < TB; ++b) gsum += gpart[(size_t)b * E + e];
  const float me = gsum / (float)T;
  const float ce = (float)r1 / (float)T;
  part[e] = me * ce;
  __syncthreads();
  if (e == 0) {
    float tot = 0.f;
    for (int i = 0; i < E; ++i) tot += part[i];
    out_laux[0] = tot * (float)E;   // mean(me*ce)*E*E == sum*E
  }
}

// ---------------------------------------------------------------- rank, capacity, scatter
__global__ __launch_bounds__(CHUNK) void scatter_kernel(const int* __restrict__ idx1,
                                                        const int* __restrict__ idx2,
                                                        const float* __restrict__ g1,
                                                        const float* __restrict__ g2,
                                                        const int* __restrict__ off1,
                                                        const int* __restrict__ off2,
                                                        const int* __restrict__ count1,
                                                        float* __restrict__ out) {
  __shared__ int wc1[8][E];
  __shared__ int wc2[8][E];
  const int lane = threadIdx.x & 31;
  const int wv   = threadIdx.x >> 5;
  for (int i = threadIdx.x; i < 8 * E; i += CHUNK) {
    (&wc1[0][0])[i] = 0; (&wc2[0][0])[i] = 0;
  }
  __syncthreads();

  const int t  = blockIdx.x * CHUNK + threadIdx.x;
  const int m1 = idx1[t], m2 = idx2[t];

  // ordered intra-wave rank among earlier lanes with same expert
  int r1 = 0, r2 = 0;
#pragma unroll
  for (int j = 0; j < 32; ++j) {
    const int v1 = __shfl(m1, j, 32);
    const int v2 = __shfl(m2, j, 32);
    if (j < lane) { r1 += (v1 == m1); r2 += (v2 == m2); }
  }
  atomicAdd(&wc1[wv][m1], 1);
  atomicAdd(&wc2[wv][m2], 1);
  __syncthreads();
  // exclusive prefix of wave counts per expert
  if (threadIdx.x < E) {
    const int e = threadIdx.x;
    int s1 = 0, s2 = 0;
    for (int w2 = 0; w2 < 8; ++w2) {
      int v = wc1[w2][e]; wc1[w2][e] = s1; s1 += v;
      int u = wc2[w2][e]; wc2[w2][e] = s2; s2 += u;
    }
  }
  __syncthreads();

  const int loc1 = off1[blockIdx.x * E + m1] + wc1[wv][m1] + r1;
  const int loc2 = off2[blockIdx.x * E + m2] + wc2[wv][m2] + r2 + count1[m2];
  const bool k1 = loc1 < CAP;
  const bool k2 = loc2 < CAP;
  const float a1 = k1 ? g1[t] : 0.f;
  const float a2 = k2 ? g2[t] : 0.f;
  const float denom = fmaxf(a1 + a2, EPSF);
  const float w1 = a1 / denom, w2 = a2 / denom;

  const size_t TEC = (size_t)T * E * CAP;
  float* combine  = out + 1;
  float* dispatch = out + 1 + TEC;
  if (k1) {
    const size_t o = ((size_t)t * E + m1) * CAP + (size_t)loc1;
    combine[o]  = w1;
    dispatch[o] = (w1 != 0.f) ? 1.f : 0.f;
  }
  if (k2) {
    const size_t o = ((size_t)t * E + m2) * CAP + (size_t)loc2;
    combine[o]  = w2;
    dispatch[o] = (w2 != 0.f) ? 1.f : 0.f;
  }
}

// ---------------------------------------------------------------- launcher
extern "C" void kernel_launch(void* const* d_in, const int* in_sizes, int n_in,
                              void* d_out, int out_size, void* d_ws, size_t ws_size,
                              hipStream_t stream) {
  const float* x  = (const float*)d_in[0];
  const float* wg = (const float*)d_in[1];
  const float* gn = (const float*)d_in[2];
  float* out = (float*)d_out;

  char* p = (char*)d_ws;
  float* logits    = (float*)p; p += sizeof(float) * (size_t)T * E;
  int*   idx1      = (int*)p;   p += sizeof(int) * T;
  int*   idx2      = (int*)p;   p += sizeof(int) * T;
  float* g1        = (float*)p; p += sizeof(float) * T;
  float* g2        = (float*)p; p += sizeof(float) * T;
  float* gpart     = (float*)p; p += sizeof(float) * (size_t)TB * E;
  int*   hist1     = (int*)p;   p += sizeof(int) * NB * E;
  int*   hist2     = (int*)p;   p += sizeof(int) * NB * E;
  int*   off1      = (int*)p;   p += sizeof(int) * NB * E;
  int*   off2      = (int*)p;   p += sizeof(int) * NB * E;
  int*   count1    = (int*)p;   p += sizeof(int) * E;

  const long long total = 1LL + 2LL * T * E * CAP;  // l_aux + combine + dispatch
  const long long n4 = total >> 2;
  const int ntail = (int)(total - n4 * 4);

  fill_zero4<<<4096, 256, 0, stream>>>((v4f*)d_out, n4, out + n4 * 4, ntail);
  gemm_logits<<<T / 64, 128, 0, stream>>>(x, wg, logits);
  top2_gate<<<TB, 256, 0, stream>>>(logits, gn, idx1, idx2, g1, g2, gpart);
  hist_kernel<<<NB, CHUNK, 0, stream>>>(idx1, idx2, hist1, hist2);
  scan_kernel<<<1, 64, 0, stream>>>(hist1, hist2, off1, off2, count1, gpart, out);
  scatter_kernel<<<NB, CHUNK, 0, stream>>>(idx1, idx2, g1, g2, off1, off2, count1, out);
}